// PoseEncoder_67697274520113
// MI455X (gfx1250) — compile-verified
//
#include <hip/hip_runtime.h>
#include <math.h>

typedef __attribute__((ext_vector_type(2))) float v2f;
typedef __attribute__((ext_vector_type(8))) float v8f;

#define NJ 24
#define GS 68     // LDS row stride for G (A rows): bank = 4*row + k -> conflict-free
#define BPS 160   // dword stride per k-pair of swizzled B: +160 -> +32 banks,
                  // so the two lane-halves (pair p / p+1) hit disjoint banks

// ---------------------------------------------------------------------------
// Kernel 1: pose MLP + build B matrix [64 x 64] = [P(24) ; Wl[64:103] ; 0]
// P[j] = relu(embed(pose_j) @ W1 + b1) @ W2 + b2, then @ Wl[:64] + bl
// ---------------------------------------------------------------------------
__global__ void pose_prep_kernel(const float* __restrict__ pose,
                                 const float* __restrict__ W1, const float* __restrict__ b1,
                                 const float* __restrict__ W2, const float* __restrict__ b2,
                                 const float* __restrict__ Wl, const float* __restrict__ bl,
                                 float* __restrict__ Bws, int* __restrict__ flags) {
    __shared__ float pe[NJ * 30];
    __shared__ float gbuf[NJ * 128];
    __shared__ float fp[NJ * 64];
    const int t = threadIdx.x;  // 256 threads
    if (t == 0) *flags = 0;
    if (t < NJ) {
        float x[6];
        #pragma unroll
        for (int d = 0; d < 6; ++d) { x[d] = pose[t * 6 + d]; pe[t * 30 + d] = x[d]; }
        float f = 1.0f;
        #pragma unroll
        for (int k = 0; k < 2; ++k) {
            #pragma unroll
            for (int d = 0; d < 6; ++d) {
                float s, c; sincosf(f * x[d], &s, &c);
                pe[t * 30 + 6 + k * 12 + d] = s;
                pe[t * 30 + 6 + k * 12 + 6 + d] = c;
            }
            f *= 2.0f;
        }
    }
    __syncthreads();
    for (int i = t; i < NJ * 128; i += 256) {
        int j = i >> 7, h = i & 127;
        float s = b1[h];
        for (int q = 0; q < 30; ++q) s += pe[j * 30 + q] * W1[q * 128 + h];
        gbuf[i] = fmaxf(s, 0.0f);
    }
    __syncthreads();
    for (int i = t; i < NJ * 64; i += 256) {
        int j = i >> 6, c = i & 63;
        float s = b2[c];
        for (int h = 0; h < 128; ++h) s += gbuf[j * 128 + h] * W2[h * 64 + c];
        fp[i] = s;
    }
    __syncthreads();
    for (int i = t; i < 64 * 64; i += 256) {
        int k = i >> 6, c = i & 63;
        float v;
        if (k < NJ) {              // P rows (includes bl)
            v = bl[c];
            for (int h = 0; h < 64; ++h) v += fp[k * 64 + h] * Wl[h * 64 + c];
        } else if (k < 63) {       // x_e part of Wl
            v = Wl[(64 + k - NJ) * 64 + c];
        } else {
            v = 0.0f;              // K padding row
        }
        Bws[i] = v;
    }
}

// ---------------------------------------------------------------------------
// Kernel 2: per-point window/embed reduction into LDS, then fp32 WMMA GEMM.
// blockDim = 128 (4 waves), each block handles 128 points.
// B is staged into LDS pre-swizzled as [k-pair][col][2] so each lane's WMMA
// B operand (B[ka][col], B[ka+1][col]) is one aligned 8-byte LDS load into an
// adjacent VGPR pair -> no repack movs in front of v_wmma.
// ---------------------------------------------------------------------------
__global__ void __launch_bounds__(128, 1)
pose_main_kernel(const float* __restrict__ pts,
                 const float* __restrict__ w2l,
                 const float* __restrict__ axis_scale,
                 const float* __restrict__ Bws,
                 float* __restrict__ outMask,
                 float* __restrict__ outLocal,
                 int* __restrict__ flags) {
    __shared__ float Gsh[128 * GS];
    __shared__ float Bsw[32 * BPS];
    const int t = threadIdx.x;
    const int n0 = blockIdx.x * 128;
    const int n = n0 + t;

    // stage B into LDS, swizzled: element (k, c) -> Bsw[(k/2)*BPS + c*2 + (k&1)]
    for (int i = t; i < 64 * 64; i += 128) {
        const int k = i >> 6, c = i & 63;
        Bsw[(k >> 1) * BPS + c * 2 + (k & 1)] = Bws[i];
    }

    // ---- phase 1: build this point's G row: [window(24) | sum_j w*x_e (39) | 0]
    const float px = pts[n * 3 + 0];
    const float py = pts[n * 3 + 1];
    const float pz = pts[n * 3 + 2];
    float acc[39];
    #pragma unroll
    for (int i = 0; i < 39; ++i) acc[i] = 0.0f;
    int jmask = 0;
    float* grow = &Gsh[t * GS];

    #pragma unroll 1
    for (int j = 0; j < NJ; ++j) {
        const float* M = w2l + j * 16;
        float xl[3];
        xl[0] = M[0] * px + M[1] * py + M[2]  * pz + M[3];
        xl[1] = M[4] * px + M[5] * py + M[6]  * pz + M[7];
        xl[2] = M[8] * px + M[9] * py + M[10] * pz + M[11];
        xl[0] /= fabsf(axis_scale[j * 3 + 0]);
        xl[1] /= fabsf(axis_scale[j * 3 + 1]);
        xl[2] /= fabsf(axis_scale[j * 3 + 2]);

        float s6 = 0.0f;
        #pragma unroll
        for (int d = 0; d < 3; ++d) { float u = xl[d] * xl[d]; s6 += u * u * u; }
        const float win = expf(-2.0f * s6);
        grow[j] = win;

        if (fabsf(xl[0]) <= 1.0f && fabsf(xl[1]) <= 1.0f && fabsf(xl[2]) <= 1.0f)
            jmask |= 1 << j;

        acc[0] += win * xl[0];
        acc[1] += win * xl[1];
        acc[2] += win * xl[2];
        float f = 1.0f;
        #pragma unroll
        for (int k = 0; k < 6; ++k) {
            #pragma unroll
            for (int d = 0; d < 3; ++d) {
                float sv, cv; sincosf(f * xl[d], &sv, &cv);
                acc[3 + k * 6 + d]     += win * sv;
                acc[3 + k * 6 + 3 + d] += win * cv;
            }
            f *= 2.0f;
        }
    }
    #pragma unroll
    for (int i = 0; i < 39; ++i) grow[24 + i] = acc[i];
    grow[63] = 0.0f;

    outMask[n] = jmask ? 1.0f : 0.0f;
    if (jmask) atomicOr(flags, jmask);
    __syncthreads();

    // ---- phase 2: G[128,64] @ B[64,64] via v_wmma_f32_16x16x4_f32
    const int wave = t >> 5;
    const int lane = t & 31;
    const int mcol = lane & 15;          // M index for A, N index for B/C
    const int hi   = (lane >= 16);

    #pragma unroll 1
    for (int mt = 0; mt < 2; ++mt) {
        const int mbase = wave * 32 + mt * 16;   // row block inside this workgroup
        const float* arow = &Gsh[(mbase + mcol) * GS];
        #pragma unroll 1
        for (int cp = 0; cp < 2; ++cp) {         // pair of 16-wide N tiles
            v8f c0 = v8f{};
            v8f c1 = v8f{};
            const int col0 = cp * 32 + mcol;
            #pragma unroll
            for (int k0 = 0; k0 < 64; k0 += 4) {
                const int ka = k0 + (hi ? 2 : 0);       // even
                const int pb = (k0 >> 1) + hi;          // k-pair index
                v2f a  = *(const v2f*)&arow[ka];
                v2f b0 = *(const v2f*)&Bsw[pb * BPS + col0 * 2];
                v2f b1 = *(const v2f*)&Bsw[pb * BPS + (col0 + 16) * 2];
                c0 = __builtin_amdgcn_wmma_f32_16x16x4_f32(
                    false, a, false, b0, (short)0, c0, false, false);
                c1 = __builtin_amdgcn_wmma_f32_16x16x4_f32(
                    false, a, false, b1, (short)0, c1, false, false);
            }
            // C/D layout: VGPR r -> row r (lanes 0-15) / r+8 (lanes 16-31)
            const int rbase = n0 + mbase + (hi ? 8 : 0);
            #pragma unroll
            for (int r = 0; r < 8; ++r) {
                outLocal[(rbase + r) * 64 + cp * 32 + mcol]      = c0[r];
                outLocal[(rbase + r) * 64 + cp * 32 + 16 + mcol] = c1[r];
            }
        }
    }
}

// ---------------------------------------------------------------------------
// Kernel 3: valid_jts bitmask -> floats
// ---------------------------------------------------------------------------
__global__ void jts_kernel(const int* __restrict__ flags, float* __restrict__ out) {
    const int j = threadIdx.x;
    if (j < NJ) out[j] = ((*flags >> j) & 1) ? 1.0f : 0.0f;
}

extern "C" void kernel_launch(void* const* d_in, const int* in_sizes, int n_in,
                              void* d_out, int out_size, void* d_ws, size_t ws_size,
                              hipStream_t stream) {
    const float* pts        = (const float*)d_in[0];
    const float* pose       = (const float*)d_in[1];
    const float* w2l        = (const float*)d_in[2];
    const float* axis_scale = (const float*)d_in[3];
    const float* W1         = (const float*)d_in[4];
    const float* b1         = (const float*)d_in[5];
    const float* W2         = (const float*)d_in[6];
    const float* b2         = (const float*)d_in[7];
    const float* Wl         = (const float*)d_in[8];
    const float* bl         = (const float*)d_in[9];

    const int N = in_sizes[0] / 3;   // 65536

    int*   flags = (int*)d_ws;
    float* Bws   = (float*)d_ws + 64;   // 64x64 B matrix, 256B offset

    float* outMask  = (float*)d_out;           // [N]
    float* outJts   = outMask + N;             // [24]
    float* outLocal = outJts + NJ;             // [N,64]

    pose_prep_kernel<<<1, 256, 0, stream>>>(pose, W1, b1, W2, b2, Wl, bl, Bws, flags);
    pose_main_kernel<<<N / 128, 128, 0, stream>>>(pts, w2l, axis_scale, Bws,
                                                  outMask, outLocal, flags);
    jts_kernel<<<1, 32, 0, stream>>>(flags, outJts);
}